// MoEGate_81432579932347
// MI455X (gfx1250) — compile-verified
//
#include <hip/hip_runtime.h>
#include <hip/hip_bf16.h>

typedef __attribute__((ext_vector_type(2))) float v2f;
typedef __attribute__((ext_vector_type(8))) float v8f;

#define N_TOK   65536   // BSZ * SEQ
#define HDIM    512
#define NEXP    4
#define SEQLEN  8192
#define BSZ     8
#define TOPK    2
#define ALPHA   0.1f

// ---------------------------------------------------------------------------
// ws layout (floats): [0 .. 31]  = sum over tokens of scores[b][e]
//                     [32 .. 63] = count of expert e chosen in top-k for batch b
// ---------------------------------------------------------------------------

__global__ void moe_gate_zero_ws(float* __restrict__ ws) {
    ws[threadIdx.x] = 0.0f;   // 64 threads
}

__global__ __launch_bounds__(256) void moe_gate_main(
    const float* __restrict__ x,   // [N_TOK, HDIM]
    const float* __restrict__ w,   // [NEXP, HDIM]
    float* __restrict__ out,       // [4*N_TOK + 1]
    float* __restrict__ ws)        // [64]
{
    // Weights zero-padded to 16 "expert" rows so the B fragment needs no mask.
    __shared__ float lds_w[16 * HDIM];        // 32 KB
    __shared__ float lds_logits[128 * NEXP];  // 2 KB
    __shared__ float lds_sum[NEXP];
    __shared__ float lds_cnt[NEXP];

    const int tid = threadIdx.x;

    for (int i = tid; i < 16 * HDIM; i += 256)
        lds_w[i] = (i < NEXP * HDIM) ? w[i] : 0.0f;
    if (tid < NEXP) { lds_sum[tid] = 0.0f; lds_cnt[tid] = 0.0f; }
    __syncthreads();

    const int wave  = tid >> 5;
    const int lane  = tid & 31;
    const int row   = lane & 15;   // M for A-frag, N for B-frag
    const int khalf = lane >> 4;   // 0 -> K=0,1 ; 1 -> K=2,3
    const int blockBase = blockIdx.x * 128;
    const int tokBase   = blockBase + wave * 16;

    // A fragment source: x[tokBase+row, h + 2*khalf .. +1], streamed once.
    const float* xrow = x + (size_t)(tokBase + row) * HDIM + 2 * khalf;
    // B fragment source: lds_w[row, h + 2*khalf .. +1] (row >= 4 reads zeros).
    const float* wrow = lds_w + row * HDIM + 2 * khalf;

    v8f acc = {};
#pragma unroll 4
    for (int h = 0; h < HDIM; h += 4) {
        v2f a = __builtin_nontemporal_load((const v2f*)(xrow + h));
        v2f b = *(const v2f*)(wrow + h);
        // D(16x16 f32) += A(16x4 f32) x B(4x16 f32)
        acc = __builtin_amdgcn_wmma_f32_16x16x4_f32(
                  false, a, false, b, (short)0, acc, false, false);
    }

    // Scatter the 4 useful expert columns of D to LDS.
    // D layout: VGPR j, lanes 0-15 -> M=j, N=lane; lanes 16-31 -> M=j+8, N=lane-16.
    if (row < NEXP) {
#pragma unroll
        for (int j = 0; j < 8; ++j)
            lds_logits[(wave * 16 + khalf * 8 + j) * NEXP + row] = acc[j];
    }
    __syncthreads();

    // Per-token softmax / top-2 / outputs / aux accumulation.
    if (tid < 128) {
        const int t = blockBase + tid;
        const float l0 = lds_logits[tid * NEXP + 0];
        const float l1 = lds_logits[tid * NEXP + 1];
        const float l2 = lds_logits[tid * NEXP + 2];
        const float l3 = lds_logits[tid * NEXP + 3];

        const float m  = fmaxf(fmaxf(l0, l1), fmaxf(l2, l3));
        const float e0 = __expf(l0 - m);
        const float e1 = __expf(l1 - m);
        const float e2 = __expf(l2 - m);
        const float e3 = __expf(l3 - m);
        const float inv = 1.0f / (e0 + e1 + e2 + e3);
        const float p0 = e0 * inv, p1 = e1 * inv, p2 = e2 * inv, p3 = e3 * inv;

        // top-1 (strict > keeps lowest index on ties, matching lax.top_k)
        int i0 = 0; float b0 = l0;
        if (l1 > b0) { b0 = l1; i0 = 1; }
        if (l2 > b0) { b0 = l2; i0 = 2; }
        if (l3 > b0) { b0 = l3; i0 = 3; }
        // top-2
        int i1 = -1; float b1 = 0.0f;
        if (i0 != 0)              { b1 = l0; i1 = 0; }
        if (i0 != 1 && (i1 < 0 || l1 > b1)) { b1 = l1; i1 = 1; }
        if (i0 != 2 && (i1 < 0 || l2 > b1)) { b1 = l2; i1 = 2; }
        if (i0 != 3 && (i1 < 0 || l3 > b1)) { b1 = l3; i1 = 3; }

        const float pa = (i0 == 0) ? p0 : (i0 == 1) ? p1 : (i0 == 2) ? p2 : p3;
        const float pb = (i1 == 0) ? p0 : (i1 == 1) ? p1 : (i1 == 2) ? p2 : p3;
        const float denom = pa + pb + 1e-20f;

        out[(size_t)t * 2 + 0] = (float)i0;
        out[(size_t)t * 2 + 1] = (float)i1;
        out[(size_t)(2 * N_TOK) + (size_t)t * 2 + 0] = pa / denom;
        out[(size_t)(2 * N_TOK) + (size_t)t * 2 + 1] = pb / denom;

        atomicAdd(&lds_sum[0], p0);
        atomicAdd(&lds_sum[1], p1);
        atomicAdd(&lds_sum[2], p2);
        atomicAdd(&lds_sum[3], p3);
        atomicAdd(&lds_cnt[i0], 1.0f);
        atomicAdd(&lds_cnt[i1], 1.0f);
    }
    __syncthreads();

    if (tid < NEXP) {
        const int b = blockBase >> 13;   // 8192 tokens per batch row; 128 | 8192
        atomicAdd(&ws[b * NEXP + tid],            lds_sum[tid]);
        atomicAdd(&ws[BSZ * NEXP + b * NEXP + tid], lds_cnt[tid]);
    }
}

__global__ void moe_gate_aux(const float* __restrict__ ws, float* __restrict__ out) {
    if (threadIdx.x == 0) {
        float aux = 0.0f;
        for (int b = 0; b < BSZ; ++b)
            for (int e = 0; e < NEXP; ++e) {
                const float ce = ws[BSZ * NEXP + b * NEXP + e] *
                                 ((float)NEXP / ((float)SEQLEN * (float)TOPK));
                const float msc = ws[b * NEXP + e] / (float)SEQLEN;
                aux += ce * msc;
            }
        out[(size_t)4 * N_TOK] = aux * (ALPHA / (float)BSZ);
    }
}

extern "C" void kernel_launch(void* const* d_in, const int* in_sizes, int n_in,
                              void* d_out, int out_size, void* d_ws, size_t ws_size,
                              hipStream_t stream) {
    const float* x = (const float*)d_in[0];   // hidden_states [8,8192,512] f32
    const float* w = (const float*)d_in[1];   // weight [4,512] f32
    float* out = (float*)d_out;               // [2T idx][2T weights][1 aux]
    float* ws  = (float*)d_ws;

    moe_gate_zero_ws<<<1, 64, 0, stream>>>(ws);
    moe_gate_main<<<N_TOK / 128, 256, 0, stream>>>(x, w, out, ws);
    moe_gate_aux<<<1, 32, 0, stream>>>(ws, out);
}